// _Encoder_16612933500997
// MI455X (gfx1250) — compile-verified
//
#include <hip/hip_runtime.h>

// One wave (32 lanes) per batch element.
//   1) async-load 13x16 f32 tile -> LDS (global_load_async_to_lds_b128, ASYNCcnt)
//   2) build A-layout regs of M_A (16x28 real form of the 8x13 complex matrix)
//   3) M = M_A * M_A^T  via 7x  v_wmma_f32_16x16x4_f32  (symmetric-operand trick)
//   4) Frobenius-normalize, then 4 repeated squarings (4 WMMA each) -> M^16
//   5) y = M^16 * ones  via 4 WMMA  (dominant eigenvector of A A^H, real form)
//   6) per-lane post-processing (argmax / phase fix / quantize) + 27-bit store

#define WAVES_PER_BLOCK 8

typedef float v2f __attribute__((ext_vector_type(2)));
typedef float v8f __attribute__((ext_vector_type(8)));

__device__ __forceinline__ v8f wmma_16x16x4(float a0, float a1, float b0, float b1, v8f c) {
  v2f A; A.x = a0; A.y = a1;
  v2f B; B.x = b0; B.y = b1;
  // 8 args: (neg_a, A, neg_b, B, c_mod, C, reuse_a, reuse_b)
  return __builtin_amdgcn_wmma_f32_16x16x4_f32(false, A, false, B, (short)0, c, false, false);
}

// Element of M_A (16 x 28):  [ Ar  -Ai  0 ]      Ar[m][j] = x[j][2m]
//                            [ Ai   Ar  0 ]      Ai[m][j] = x[j][2m+1],  j = 0..12
// Branchless: unconditional LDS load (clamped), selects for sign/zero.
__device__ __forceinline__ float loadMA(const float* xs, int r, int k) {
  const int  valid  = (k < 26);
  const int  kk     = valid ? k : 0;
  const int  second = (kk >= 13) ? 1 : 0;
  const int  row    = second ? (kk - 13) : kk;   // 0..12
  const int  rr     = r & 7;
  const int  rhi    = (r >= 8) ? 1 : 0;
  const int  col    = 2 * rr + (rhi ^ second);   // parity: {Ar even, Ai odd}
  float v = xs[row * 16 + col];
  v = (second && !rhi) ? -v : v;                 // -Ai block
  return valid ? v : 0.0f;
}

// C/D layout -> A-layout via symmetry (M[r][k] == M[k][r]), with scale s.
__device__ __forceinline__ void cToA(v8f c, bool hi, int r, float s, float* ta) {
  float e0 = c[0], e1 = c[1], e2 = c[2], e3 = c[3];
  float e4 = c[4], e5 = c[5], e6 = c[6], e7 = c[7];
  float a0 = __shfl(e0, r),      a1 = __shfl(e1, r);
  float a2 = __shfl(e2, r),      a3 = __shfl(e3, r);
  float a4 = __shfl(e4, r),      a5 = __shfl(e5, r);
  float a6 = __shfl(e6, r),      a7 = __shfl(e7, r);
  float b0 = __shfl(e0, r + 16), b1 = __shfl(e1, r + 16);
  float b2 = __shfl(e2, r + 16), b3 = __shfl(e3, r + 16);
  float b4 = __shfl(e4, r + 16), b5 = __shfl(e5, r + 16);
  float b6 = __shfl(e6, r + 16), b7 = __shfl(e7, r + 16);
  ta[0] = s * (hi ? a2 : a0);  ta[1] = s * (hi ? a3 : a1);
  ta[2] = s * (hi ? a6 : a4);  ta[3] = s * (hi ? a7 : a5);
  ta[4] = s * (hi ? b2 : b0);  ta[5] = s * (hi ? b3 : b1);
  ta[6] = s * (hi ? b6 : b4);  ta[7] = s * (hi ? b7 : b5);
}

__device__ __forceinline__ float pick8(const float* v, int idx) {
  float r = v[0];
#pragma unroll
  for (int m = 1; m < 8; ++m) r = (idx == m) ? v[m] : r;
  return r;
}

__global__ __launch_bounds__(WAVES_PER_BLOCK * 32)
void enc_kernel(const float* __restrict__ x, float* __restrict__ out, int nBatch) {
  __shared__ float lds[WAVES_PER_BLOCK * 13 * 16];
  const int lane = threadIdx.x & 31;
  const int wave = threadIdx.x >> 5;
  const int b    = blockIdx.x * WAVES_PER_BLOCK + wave;
  if (b >= nBatch) return;

  float* xs = &lds[wave * 208];                  // 832 B, 16B-aligned tile
  const float* gbase = x + (size_t)b * 208;

  // ---- async direct-to-LDS tile load: 832 B per wave, 2x B128 issues ----
  // LDS generic pointers carry the allocation-relative byte offset in the
  // low 32 bits -> that is exactly the VDST (LDS address) operand.
  {
    const unsigned ldsb = (unsigned)(uintptr_t)(void*)xs;
    unsigned voff0 = (unsigned)(lane * 16);      // 32 lanes x 16B = 512 B
    asm volatile("global_load_async_to_lds_b128 %0, %1, %2 offset:0"
                 :: "v"(ldsb + voff0), "v"(voff0), "s"(gbase) : "memory");
    if (lane < 20) {                             // 20 lanes x 16B = 320 B
      unsigned voff1 = (unsigned)(512 + lane * 16);
      asm volatile("global_load_async_to_lds_b128 %0, %1, %2 offset:0"
                   :: "v"(ldsb + voff1), "v"(voff1), "s"(gbase) : "memory");
    }
    asm volatile("s_wait_asynccnt 0x0" ::: "memory");
  }
  __syncthreads();

  const int  r  = lane & 15;
  const bool hi = lane >= 16;

  // ---- A-layout operand regs of M_A (also valid as the B operand) ----
  float ma[14];
#pragma unroll
  for (int c = 0; c < 7; ++c) {
    int k0 = 4 * c + (hi ? 2 : 0);
    ma[2 * c]     = loadMA(xs, r, k0);
    ma[2 * c + 1] = loadMA(xs, r, k0 + 1);
  }

  // ---- M = M_A * M_A^T : 7 WMMA, K accumulation ----
  v8f acc = {0.f, 0.f, 0.f, 0.f, 0.f, 0.f, 0.f, 0.f};
#pragma unroll
  for (int c = 0; c < 7; ++c)
    acc = wmma_16x16x4(ma[2 * c], ma[2 * c + 1], ma[2 * c], ma[2 * c + 1], acc);

  // ---- Frobenius normalization (so M^16 stays in f32 range) ----
  float fro = 0.f;
#pragma unroll
  for (int j = 0; j < 8; ++j) { float e = acc[j]; fro += e * e; }
#pragma unroll
  for (int s = 1; s < 32; s <<= 1) fro += __shfl_xor(fro, s);
  const float scal = rsqrtf(fro + 1e-37f);

  float ta[8];
  cToA(acc, hi, r, scal, ta);

  // ---- 4 repeated squarings: M -> M^16 (4 WMMA each) ----
#pragma unroll
  for (int it = 0; it < 4; ++it) {
    v8f m2 = {0.f, 0.f, 0.f, 0.f, 0.f, 0.f, 0.f, 0.f};
#pragma unroll
    for (int c4 = 0; c4 < 4; ++c4)
      m2 = wmma_16x16x4(ta[2 * c4], ta[2 * c4 + 1], ta[2 * c4], ta[2 * c4 + 1], m2);
    cToA(m2, hi, r, 1.0f, ta);
  }

  // ---- y = M^16 * ones : dominant eigenvector (real form), 4 WMMA ----
  v8f y = {0.f, 0.f, 0.f, 0.f, 0.f, 0.f, 0.f, 0.f};
#pragma unroll
  for (int c4 = 0; c4 < 4; ++c4)
    y = wmma_16x16x4(ta[2 * c4], ta[2 * c4 + 1], 1.0f, 1.0f, y);

  // ---- broadcast full 16-vector to every lane ----
  float vr[8], vi[8];
#pragma unroll
  for (int m = 0; m < 8; ++m) { float e = y[m]; vr[m] = __shfl(e, 0); vi[m] = __shfl(e, 16); }

  // ---- argmax of |v|^2 (first occurrence) ----
  int   idx  = 0;
  float best = vr[0] * vr[0] + vi[0] * vi[0];
#pragma unroll
  for (int m = 1; m < 8; ++m) {
    float mg = vr[m] * vr[m] + vi[m] * vi[m];
    if (mg > best) { best = mg; idx = m; }
  }

  // ---- phase fix: p = v * conj(v[idx]); normalize by max |.| ----
  const float cr = pick8(vr, idx);
  const float ci = -pick8(vi, idx);
  float pr[8], pi[8], mx = 0.f;
#pragma unroll
  for (int m = 0; m < 8; ++m) {
    pr[m] = vr[m] * cr - vi[m] * ci;
    pi[m] = vr[m] * ci + vi[m] * cr;
    mx = fmaxf(mx, fmaxf(fabsf(pr[m]), fabsf(pi[m])));
  }
  const float inv = 1.0f / mx;
#pragma unroll
  for (int m = 0; m < 8; ++m) { pr[m] *= inv; pi[m] *= inv; }

  // ---- permutation: indices != idx in increasing order (first 7) ----
  float prp[7], pip[7];
#pragma unroll
  for (int t = 0; t < 7; ++t) {
    int j = (t < idx) ? t : t + 1;
    prp[t] = pick8(pr, j);
    pip[t] = pick8(pi, j);
  }

  // ---- quantize ----
  float v1b[4];
#pragma unroll
  for (int i = 0; i < 4; ++i) v1b[i] = rintf((prp[i] + 1.0f) * 1.5f);   // 0..3
  float s3 = 0.f, p3 = 1.0f;
#pragma unroll
  for (int i = 0; i < 10; ++i) {                                        // trits 0..2
    float vc = (i < 3) ? prp[4 + i] : pip[i - 3];
    s3 += rintf(vc + 1.0f) * p3;
    p3 *= 3.0f;
  }

  // ---- 27 output bits, one lane per bit (coalesced) ----
  if (lane < 27) {
    int iv;
    if (lane < 3) {
      iv = idx >> lane;
    } else if (lane < 11) {
      int q = lane - 3, i = q >> 1, a = q & 1;
      float v = v1b[0];
      v = (i == 1) ? v1b[1] : v;
      v = (i == 2) ? v1b[2] : v;
      v = (i == 3) ? v1b[3] : v;
      iv = ((int)v) >> a;
    } else {
      iv = ((int)s3) >> (lane - 11);
    }
    out[(size_t)b * 27 + lane] = (float)(iv & 1);
  }
}

extern "C" void kernel_launch(void* const* d_in, const int* in_sizes, int n_in,
                              void* d_out, int out_size, void* d_ws, size_t ws_size,
                              hipStream_t stream) {
  const float* x = (const float*)d_in[0];
  float* out = (float*)d_out;
  const int nBatch = in_sizes[0] / (13 * 16);   // 131072
  const int blocks = (nBatch + WAVES_PER_BLOCK - 1) / WAVES_PER_BLOCK;
  enc_kernel<<<blocks, WAVES_PER_BLOCK * 32, 0, stream>>>(x, out, nBatch);
}